// WeightedSumQuatEmbedding_26654567039200
// MI455X (gfx1250) — compile-verified
//
#include <hip/hip_runtime.h>

// ---------------- problem constants (fixed by the reference) ----------------
constexpr int F_FIELDS  = 20;
constexpr int FIELD_DIM = 50000;
constexpr int A_ACT     = 4;      // len(ACTION)
constexpr int M_SUB     = 4;      // M
constexpr int PLEN      = 16;     // EMBED_DIM / M
constexpr int EMB       = 64;
constexpr int MAXK      = 512;
constexpr long long V_TOT = (long long)F_FIELDS * FIELD_DIM;  // 1,000,000
constexpr int TB        = 64;     // batch rows per block
constexpr int NTHREADS  = 256;    // 8 wave32s; 4 lanes (one per m) per batch row
constexpr int SLAB_BYTES = MAXK * EMB * 4;  // 128 KB per-field codebook slab

typedef float v4f __attribute__((ext_vector_type(4)));
// Exact GCC-vector type the async-LDS builtin expects pointers to.
typedef int gv4i __attribute__((vector_size(16)));
typedef __attribute__((address_space(1))) gv4i* gv4i_g;   // global
typedef __attribute__((address_space(3))) gv4i* gv4i_l;   // LDS

#define ASYNC_LDS_OK  (__has_builtin(__builtin_amdgcn_global_load_async_to_lds_b128))
#define WAIT_ASYNC_OK (__has_builtin(__builtin_amdgcn_s_wait_asynccnt))

// 16B global -> LDS copy. CDNA5 path: GLOBAL_LOAD_ASYNC_TO_LDS_B128 (per-lane
// scatter, tracked by ASYNCcnt). Fallback: copy through VGPRs.
__device__ __forceinline__ void async_cp16(const void* g, void* l) {
#if ASYNC_LDS_OK
  __builtin_amdgcn_global_load_async_to_lds_b128((gv4i_g)(g), (gv4i_l)(l), 0, 0);
#else
  *reinterpret_cast<v4f*>(l) = *reinterpret_cast<const v4f*>(g);
#endif
}

__device__ __forceinline__ void wait_async_copies() {
#if ASYNC_LDS_OK
#if WAIT_ASYNC_OK
  __builtin_amdgcn_s_wait_asynccnt(0);
#else
  asm volatile("s_wait_asynccnt 0" ::: "memory");
#endif
#endif
}

__global__ __launch_bounds__(NTHREADS) void wsq_embed_kernel(
    const int*   __restrict__ x,          // (B, 20) int32
    const float* __restrict__ arch_prob,  // (20, 4)
    const float* __restrict__ codebooks,  // (20*512, 64)
    const int*   __restrict__ cb_index,   // (4, 1e6, 4) int32
    float*       __restrict__ out,        // (B, 20, 64)
    int B)
{
  extern __shared__ float smem[];   // 512 rows x 256 B, chunk-swizzled
  char* lbase = (char*)smem;

  const int f   = blockIdx.y;
  const int tid = threadIdx.x;

  // ---- Stage field-f codebook slab (128 KB) into LDS via async DMA ----
  // Swizzle: 64B chunk m of row r lands at chunk ((m + r) & 3), so random-row
  // ds_load_b128 gathers spread over all 4 bank groups instead of 8-way
  // conflicting on one.
  {
    const char* slab = (const char*)(codebooks + (size_t)f * MAXK * EMB);
#pragma unroll
    for (int j = 0; j < SLAB_BYTES / (NTHREADS * 16); ++j) {  // 32 iters
      const int goff  = j * NTHREADS * 16 + tid * 16;  // linear byte offset
      const int row   = goff >> 8;                     // 256 B per row
      const int chunk = (goff >> 6) & 3;               // 64 B sub-chunk (= m)
      const int loff  = (row << 8) | (((chunk + row) & 3) << 6) | (goff & 48);
      async_cp16(slab + goff, lbase + loff);
    }
    wait_async_copies();
  }
  __syncthreads();

  // ---- Each quad of lanes owns one batch row b; lane-in-quad = m ----
  const int bl = tid >> 2;
  const int m  = tid & 3;
  const int b  = blockIdx.x * TB + bl;
  if (b >= B) return;

  const float4 w  = ((const float4*)arch_prob)[f];          // b128, uniform
  const float  wa[A_ACT] = {w.x, w.y, w.z, w.w};

  const int       xi = x[(size_t)b * F_FIELDS + f];
  const long long v  = (long long)xi + (long long)f * FIELD_DIM;

  // Issue all 4 random-gather index loads up front (independent, in flight
  // together). Quad lanes read 4 consecutive int32 -> one 16B segment.
  int rows[A_ACT];
#pragma unroll
  for (int a = 0; a < A_ACT; ++a)
    rows[a] = cb_index[((size_t)a * (size_t)V_TOT + (size_t)v) * M_SUB + m];

  v4f acc0 = 0.f, acc1 = 0.f, acc2 = 0.f, acc3 = 0.f;
#pragma unroll
  for (int a = 0; a < A_ACT; ++a) {
    const int row = rows[a];
    const v4f* p = (const v4f*)(lbase + (row << 8) + (((m + row) & 3) << 6));
    const float s = wa[a];
    const v4f p0 = p[0], p1 = p[1], p2 = p[2], p3 = p[3];  // 4x ds_load_b128
    acc0 += p0 * s;
    acc1 += p1 * s;
    acc2 += p2 * s;
    acc3 += p3 * s;
  }

  // Streaming 40 MB output: non-temporal stores so we don't evict the
  // L2-resident 64 MB cb_index table. 64 B contiguous per lane, 256 B per b.
  v4f* o = (v4f*)(out + (((size_t)b * F_FIELDS + f) << 6) + m * PLEN);
  __builtin_nontemporal_store(acc0, o + 0);
  __builtin_nontemporal_store(acc1, o + 1);
  __builtin_nontemporal_store(acc2, o + 2);
  __builtin_nontemporal_store(acc3, o + 3);
}

extern "C" void kernel_launch(void* const* d_in, const int* in_sizes, int n_in,
                              void* d_out, int out_size, void* d_ws, size_t ws_size,
                              hipStream_t stream) {
  const int*   x         = (const int*)d_in[0];
  const float* arch_prob = (const float*)d_in[1];
  const float* codebooks = (const float*)d_in[2];
  const int*   cb_index  = (const int*)d_in[3];
  float*       out       = (float*)d_out;

  const int B = in_sizes[0] / F_FIELDS;
  dim3 grid((B + TB - 1) / TB, F_FIELDS, 1);
  wsq_embed_kernel<<<grid, dim3(NTHREADS), SLAB_BYTES, stream>>>(
      x, arch_prob, codebooks, cb_index, out, B);
}